// SafenessLoss_63015760167697
// MI455X (gfx1250) — compile-verified
//
#include <hip/hip_runtime.h>

// SafenessLoss for MI455X (gfx1250, wave32).
//
// Bandwidth-bound: 6 x [16384,512] f32 = 192 MB @ 23.3 TB/s ~ 8.6 us floor.
// Each wave owns a 16-sample tile and a 64-wide K segment. Per 4-wide K
// chunk it forms Q = (a - e)^2 in VALU (v_pk_add neg + v_pk_mul) and runs
// the K-reduction on the XDL pipe via V_WMMA_F32_16X16X4_F32 with B = ones:
//   C[m][n] += sum_k Q[m][k]         (every column of C equals d2[m])
// A-matrix (32-bit, 16x4): lane L holds row M=L%16, K=(L/16)*2+{0,1}
//   -> contiguous 8B loads per lane, full line consumption across K loop.
// C-matrix: VGPR v on lane L holds M = v + 8*(L/16), N = L%16. Column N=0
// (lanes 0 and 16) applies the +/-1 label signs.
//
// v3 changes:
//  - KSPLIT 4 -> 8: 8192 waves for cross-wave latency hiding (~8 waves/SIMD
//    class occupancy), since the per-wave pipeline drains once per chunk.
//  - Removed single-address float atomics (8K serialized L2 RMWs would cost
//    more than the whole 8.6 us stream): block-level LDS reduction ->
//    d_ws[block], then a tiny deterministic tree-reduce kernel writes out.
//  - Dropped TH_LOAD_NT: each 128B line is reused across 8 consecutive
//    chunks in L0; NT deprioritization risks refetches, and a single-pass
//    192 MB stream gains nothing from NT anyway.

typedef __attribute__((ext_vector_type(2))) float v2f;
typedef __attribute__((ext_vector_type(8))) float v8f;

#define B_N 16384
#define D_N 512
#define KSPLIT 8
#define KSEG (D_N / KSPLIT)  // 64 floats per wave segment
#define NCHUNK (KSEG / 4)    // 16 WMMA chunks per wave
// total waves = (B_N/16) * KSPLIT = 8192 ; 8 waves/block -> 1024 blocks
#define NBLOCKS ((B_N / 16) * KSPLIT / 8)

__device__ __forceinline__ v2f ld2(const float* p, int elem_off) {
    return *(const v2f*)(p + elem_off);
}

__global__ __launch_bounds__(256) void safeness_wmma_kernel(
    const float* __restrict__ anchor,
    const float* __restrict__ e1, const float* __restrict__ e2,
    const float* __restrict__ e3, const float* __restrict__ e4,
    const float* __restrict__ e5,
    const int* __restrict__ la,
    const int* __restrict__ l1, const int* __restrict__ l2,
    const int* __restrict__ l3, const int* __restrict__ l4,
    const int* __restrict__ l5,
    float* __restrict__ ws)
{
    const int tid  = blockIdx.x * blockDim.x + threadIdx.x;
    const int wid  = tid >> 5;        // global wave id, 0..8191
    const int lane = tid & 31;

    const int tile    = wid / KSPLIT;          // 0..1023 (16-sample tile)
    const int kseg    = wid % KSPLIT;          // 0..7
    const int rowBase = tile * 16;
    const int m       = rowBase + (lane & 15); // this lane's A-matrix row
    const int koff    = (lane >> 4) * 2;       // K sub-offset per ISA layout
    const int kstart  = kseg * KSEG;

    const size_t base = (size_t)m * D_N + (size_t)(kstart + koff);
    const float* __restrict__ pa = anchor + base;
    const float* __restrict__ p1 = e1 + base;
    const float* __restrict__ p2 = e2 + base;
    const float* __restrict__ p3 = e3 + base;
    const float* __restrict__ p4 = e4 + base;
    const float* __restrict__ p5 = e5 + base;

    v2f ones; ones.x = 1.0f; ones.y = 1.0f;
    v8f acc0 = {}; v8f acc1 = {}; v8f acc2 = {}; v8f acc3 = {}; v8f acc4 = {};

    // Depth-2 pipeline buffers: slot s = chunk parity.
    v2f bufA[2], buf1[2], buf2[2], buf3[2], buf4[2], buf5[2];
    bufA[0] = ld2(pa, 0); buf1[0] = ld2(p1, 0); buf2[0] = ld2(p2, 0);
    buf3[0] = ld2(p3, 0); buf4[0] = ld2(p4, 0); buf5[0] = ld2(p5, 0);
    bufA[1] = ld2(pa, 4); buf1[1] = ld2(p1, 4); buf2[1] = ld2(p2, 4);
    buf3[1] = ld2(p3, 4); buf4[1] = ld2(p4, 4); buf5[1] = ld2(p5, 4);

#pragma unroll
    for (int c = 0; c < NCHUNK; ++c) {
        const int s = c & 1;
        const v2f a = bufA[s];
        v2f t, q;

        t = buf1[s];
        q.x = (a.x - t.x) * (a.x - t.x); q.y = (a.y - t.y) * (a.y - t.y);
        acc0 = __builtin_amdgcn_wmma_f32_16x16x4_f32(false, q, false, ones,
                                                     (short)0, acc0, false, false);
        t = buf2[s];
        q.x = (a.x - t.x) * (a.x - t.x); q.y = (a.y - t.y) * (a.y - t.y);
        acc1 = __builtin_amdgcn_wmma_f32_16x16x4_f32(false, q, false, ones,
                                                     (short)0, acc1, false, false);
        t = buf3[s];
        q.x = (a.x - t.x) * (a.x - t.x); q.y = (a.y - t.y) * (a.y - t.y);
        acc2 = __builtin_amdgcn_wmma_f32_16x16x4_f32(false, q, false, ones,
                                                     (short)0, acc2, false, false);
        t = buf4[s];
        q.x = (a.x - t.x) * (a.x - t.x); q.y = (a.y - t.y) * (a.y - t.y);
        acc3 = __builtin_amdgcn_wmma_f32_16x16x4_f32(false, q, false, ones,
                                                     (short)0, acc3, false, false);
        t = buf5[s];
        q.x = (a.x - t.x) * (a.x - t.x); q.y = (a.y - t.y) * (a.y - t.y);
        acc4 = __builtin_amdgcn_wmma_f32_16x16x4_f32(false, q, false, ones,
                                                     (short)0, acc4, false, false);

        // Prefetch chunk c+2 into this slot (WAR on the pk ops above only,
        // so these loads can issue under the WMMA chain).
        if (c + 2 < NCHUNK) {
            const int off = (c + 2) * 4;
            bufA[s] = ld2(pa, off); buf1[s] = ld2(p1, off);
            buf2[s] = ld2(p2, off); buf3[s] = ld2(p3, off);
            buf4[s] = ld2(p4, off); buf5[s] = ld2(p5, off);
        }
    }

    // Column N == 0 of the C tile lives on lanes 0 (rows 0..7 in VGPRs 0..7)
    // and 16 (rows 8..15). Apply label signs there, then reduce the block's
    // 16 half-wave partials through LDS (no global atomics).
    __shared__ float red[16];
    float partial = 0.0f;
    if ((lane & 15) == 0) {
        const int r0 = rowBase + 8 * (lane >> 4);
#pragma unroll
        for (int v = 0; v < 8; ++v) {
            const int row = r0 + v;
            const int al  = la[row];
            partial += ((l1[row] == al) ? 1.0f : -1.0f) * acc0[v];
            partial += ((l2[row] == al) ? 1.0f : -1.0f) * acc1[v];
            partial += ((l3[row] == al) ? 1.0f : -1.0f) * acc2[v];
            partial += ((l4[row] == al) ? 1.0f : -1.0f) * acc3[v];
            partial += ((l5[row] == al) ? 1.0f : -1.0f) * acc4[v];
        }
        red[((threadIdx.x >> 5) << 1) + (lane >> 4)] = partial;
    }
    __syncthreads();
    if (threadIdx.x == 0) {
        float s = 0.0f;
#pragma unroll
        for (int i = 0; i < 16; ++i) s += red[i];
        ws[blockIdx.x] = s;
    }
}

// Deterministic final reduction of the 1024 block partials.
__global__ __launch_bounds__(256) void final_reduce_kernel(
    const float* __restrict__ ws, float* __restrict__ out)
{
    __shared__ float red[256];
    float s = 0.0f;
#pragma unroll
    for (int i = 0; i < NBLOCKS / 256; ++i)
        s += ws[threadIdx.x + i * 256];
    red[threadIdx.x] = s;
    __syncthreads();
#pragma unroll
    for (int stride = 128; stride > 0; stride >>= 1) {
        if (threadIdx.x < stride) red[threadIdx.x] += red[threadIdx.x + stride];
        __syncthreads();
    }
    if (threadIdx.x == 0) out[0] = red[0] * (1.0f / (float)B_N);
}

extern "C" void kernel_launch(void* const* d_in, const int* in_sizes, int n_in,
                              void* d_out, int out_size, void* d_ws, size_t ws_size,
                              hipStream_t stream) {
    (void)in_sizes; (void)n_in; (void)out_size; (void)ws_size;

    const float* anchor = (const float*)d_in[0];
    const float* e1 = (const float*)d_in[1];
    const float* e2 = (const float*)d_in[2];
    const float* e3 = (const float*)d_in[3];
    const float* e4 = (const float*)d_in[4];
    const float* e5 = (const float*)d_in[5];
    const int* la = (const int*)d_in[6];
    const int* l1 = (const int*)d_in[7];
    const int* l2 = (const int*)d_in[8];
    const int* l3 = (const int*)d_in[9];
    const int* l4 = (const int*)d_in[10];
    const int* l5 = (const int*)d_in[11];
    float* ws  = (float*)d_ws;     // NBLOCKS floats = 4 KB of scratch
    float* out = (float*)d_out;

    safeness_wmma_kernel<<<dim3(NBLOCKS), dim3(256), 0, stream>>>(
        anchor, e1, e2, e3, e4, e5, la, l1, l2, l3, l4, l5, ws);
    final_reduce_kernel<<<dim3(1), dim3(256), 0, stream>>>(ws, out);
}